// KDEMixture_89962384982279
// MI455X (gfx1250) — compile-verified
//
#include <hip/hip_runtime.h>

typedef float v2f __attribute__((ext_vector_type(2)));
typedef float v8f __attribute__((ext_vector_type(8)));

#define KDE_EPS   1e-12f
#define KDE_BW    0.5f
#define KDE_ZF    2.0f

// ---------------------------------------------------------------------------
// Kernel 1: reduce sum of weights into d_ws[0] (single block, deterministic).
// ---------------------------------------------------------------------------
__global__ __launch_bounds__(256) void kde_wsum_kernel(const float* __restrict__ w,
                                                       float* __restrict__ out,
                                                       int n) {
    __shared__ float red[256];
    float s = 0.0f;
    for (int i = threadIdx.x; i < n; i += 256) s += w[i];
    red[threadIdx.x] = s;
    __syncthreads();
    for (int stride = 128; stride > 0; stride >>= 1) {
        if (threadIdx.x < (unsigned)stride) red[threadIdx.x] += red[threadIdx.x + stride];
        __syncthreads();
    }
    if (threadIdx.x == 0) out[0] = red[0];
}

// ---------------------------------------------------------------------------
// Kernel 2: KDE forward via V_WMMA_F32_16X16X4_F32.
// One wave32 owns a 16-row tile of query points; loops over N in 16-col tiles.
//   E(16x16) = A(16x4) * B(4x16) + C   ==  log2-space Gaussian exponent
//   km = exp2(E);  wk = km * w;  accumulate f, wk@s per-lane, butterfly-reduce.
// ---------------------------------------------------------------------------
__global__ __launch_bounds__(256) void kde_wmma_kernel(
    const float* __restrict__ points,   // [P,3]
    const float* __restrict__ sdata,    // [N,3]
    const float* __restrict__ weights,  // [N]
    const float* __restrict__ wsum_p,   // [1]
    float* __restrict__ out_f,          // [P]
    float* __restrict__ out_g,          // [P,3]
    int P, int N) {

    const int lane  = threadIdx.x & 31;
    const int wave  = threadIdx.x >> 5;
    const int tile  = blockIdx.x * 8 + wave;   // 16 rows of points per tile
    const int half  = lane >> 4;               // 0: lanes 0-15, 1: lanes 16-31
    const int ln    = lane & 15;

    const float inv_bw2 = 1.0f / (KDE_BW * KDE_BW);      // 4.0
    const float L2E     = 1.4426950408889634f;           // log2(e)
    const float c1      = L2E * inv_bw2;

    // ---- load this lane's query row (z pre-scaled), clamped for safety ----
    int prow = tile * 16 + ln;
    if (prow >= P) prow = P - 1;
    const float px = points[prow * 3 + 0];
    const float py = points[prow * 3 + 1];
    const float pz = points[prow * 3 + 2] * KDE_ZF;
    const float p2 = px * px + py * py + pz * pz;
    const float cp = -0.5f * c1 * p2;

    // ---- A operand (16x4 f32): lanes 0-15 hold K=0,1 ; lanes 16-31 K=2,3 ----
    v2f A;
    A.x = half ? pz   : px;
    A.y = half ? 1.0f : py;

    // ---- C operand: VGPR v holds row (v + 8*half), constant across N ----
    v8f Cv;
#pragma unroll
    for (int v = 0; v < 8; ++v)
        Cv[v] = __shfl(cp, v + 8 * half, 32);

    // ---- accumulators (rows v+8*half, column ln) ----
    float fA[8], gX[8], gY[8], gZ[8];
#pragma unroll
    for (int v = 0; v < 8; ++v) { fA[v] = 0.f; gX[v] = 0.f; gY[v] = 0.f; gZ[v] = 0.f; }

    // ---- main loop over data columns ----
    for (int j0 = 0; j0 < N; j0 += 16) {
        int col = j0 + ln;
        if (col >= N) col = N - 1;                 // N is a mult of 16; safety only
        const float sx = sdata[col * 3 + 0];
        const float sy = sdata[col * 3 + 1];
        const float sz = sdata[col * 3 + 2] * KDE_ZF;
        const float w  = weights[col];
        const float s2 = sx * sx + sy * sy + sz * sz;

        // B operand (4x16 f32): lanes 0-15 rows K=0,1 ; lanes 16-31 rows K=2,3
        v2f B;
        B.x = half ? (c1 * sz)          : (c1 * sx);
        B.y = half ? (-0.5f * c1 * s2)  : (c1 * sy);

        // E = A*B + C : full log2-space Gaussian exponent for the 16x16 tile
        v8f E = __builtin_amdgcn_wmma_f32_16x16x4_f32(
            /*neg_a=*/false, A, /*neg_b=*/false, B,
            /*c_mod=*/(short)0, Cv, /*reuse_a=*/false, /*reuse_b=*/false);

#pragma unroll
        for (int v = 0; v < 8; ++v) {
            const float wk = __builtin_amdgcn_exp2f(E[v]) * w;   // v_exp_f32
            fA[v] += wk;
            gX[v] += wk * sx;
            gY[v] += wk * sy;
            gZ[v] += wk * sz;
        }
    }

    // ---- butterfly-reduce over the 16 columns within each half-wave ----
#pragma unroll
    for (int v = 0; v < 8; ++v) {
#pragma unroll
        for (int m = 1; m < 16; m <<= 1) {
            fA[v] += __shfl_xor(fA[v], m, 32);
            gX[v] += __shfl_xor(gX[v], m, 32);
            gY[v] += __shfl_xor(gY[v], m, 32);
            gZ[v] += __shfl_xor(gZ[v], m, 32);
        }
    }

    // ---- finalize: normalize weights, grad, log-grad, rescale z ----
    const float winv = 1.0f / (wsum_p[0] + KDE_EPS);

#pragma unroll
    for (int v = 0; v < 8; ++v) {
        const int row = v + 8 * half;              // row within tile
        // fetch that row's query point (held by lane `row`)
        const float rpx = __shfl(px, row, 32);
        const float rpy = __shfl(py, row, 32);
        const float rpz = __shfl(pz, row, 32);

        const float f  = fA[v] * winv;
        const float Sx = gX[v] * winv;
        const float Sy = gY[v] * winv;
        const float Sz = gZ[v] * winv;

        const float g1 = inv_bw2 * (Sx - f * rpx);
        const float g2 = inv_bw2 * (Sy - f * rpy);
        const float g3 = inv_bw2 * (Sz - f * rpz);
        const float dn = 1.0f / (f + KDE_EPS);

        if (ln == 0) {                             // lane 0 -> rows 0..7, lane 16 -> rows 8..15
            const int orow = tile * 16 + row;
            if (orow < P) {
                out_f[orow]         = f;
                out_g[orow * 3 + 0] = g1 * dn;
                out_g[orow * 3 + 1] = g2 * dn;
                out_g[orow * 3 + 2] = g3 * dn * KDE_ZF;  // rescale z to original coords
            }
        }
    }
}

// ---------------------------------------------------------------------------
extern "C" void kernel_launch(void* const* d_in, const int* in_sizes, int n_in,
                              void* d_out, int out_size, void* d_ws, size_t ws_size,
                              hipStream_t stream) {
    const float* points  = (const float*)d_in[0];   // [P,3]
    const float* sdata   = (const float*)d_in[1];   // [N,3]
    const float* weights = (const float*)d_in[2];   // [N]

    const int P = in_sizes[0] / 3;
    const int N = in_sizes[2];

    float* wsum  = (float*)d_ws;
    float* out_f = (float*)d_out;        // [P]
    float* out_g = out_f + P;            // [P,3]

    kde_wsum_kernel<<<1, 256, 0, stream>>>(weights, wsum, N);

    const int tiles  = (P + 15) / 16;          // 1024 for P=16384
    const int blocks = (tiles + 7) / 8;        // 8 waves/block (wave32)
    kde_wmma_kernel<<<blocks, 256, 0, stream>>>(points, sdata, weights, wsum,
                                                out_f, out_g, P, N);
}